// GraphIsoBn_82042465288993
// MI455X (gfx1250) — compile-verified
//
#include <hip/hip_runtime.h>

#define N_NODES 100000
#define DCH     128
#define N_EDGES 1600000
#define BN_EPS  1e-5f

typedef __attribute__((ext_vector_type(2))) float v2f;
typedef __attribute__((ext_vector_type(8))) float v8f;

// ---------------------------------------------------------------------------
// Kernel A: agg = x  (fuses the "+ x_i" of GIN), zero the global BN stats.
// ---------------------------------------------------------------------------
__global__ void k_init(const float4* __restrict__ x4, float4* __restrict__ agg4,
                       float* __restrict__ gstats, int n4) {
    int i = blockIdx.x * blockDim.x + threadIdx.x;
    if (i < n4) agg4[i] = x4[i];
    if (i < 2 * DCH) gstats[i] = 0.0f;
}

// ---------------------------------------------------------------------------
// Kernel B: scatter-add. One wave per edge; each lane moves a float4
// (32 lanes x 4 = 128 channels). x and agg both fit in the 192MB L2, so
// this is L2-atomic bound; global_atomic_add_f32 x4 per lane.
// ---------------------------------------------------------------------------
__global__ void k_scatter(const float* __restrict__ x,
                          const int* __restrict__ ei,   // [2, E] flat, int32
                          float* __restrict__ agg, int nEdges) {
    int wave = (int)((blockIdx.x * blockDim.x + threadIdx.x) >> 5);
    int lane = threadIdx.x & 31;
    if (wave >= nEdges) return;
    int src = ei[wave];
    int dst = ei[nEdges + wave];
    const float4 v = *(const float4*)(x + (size_t)src * DCH + lane * 4);
    float* p = agg + (size_t)dst * DCH + lane * 4;
    atomicAdd(p + 0, v.x);
    atomicAdd(p + 1, v.y);
    atomicAdd(p + 2, v.z);
    atomicAdd(p + 3, v.w);
}

// ---------------------------------------------------------------------------
// Kernel C: fused  h2 = relu(relu(h@W1+b1)@W2+b2)  via fp32 WMMA 16x16x4,
// plus per-column partial sums / sums-of-squares for BatchNorm.
//
// 256 threads = 8 waves; workgroup covers 128 rows; each wave owns a 16-row
// stripe and all 8 column tiles of 16 (8 x v8f accumulators).
//
// LDS layout (dynamic, 132096 B):
//   [0      , 16384) floats : packed W  (float2 per (k-pair, n)) -> ds_load_b64 B-frags
//   [16384  , 32768) floats : per-wave 16x128 activation staging (layout flip D->A)
//   [32768  , 33024) floats : BN partial stats (ds_add_f32)
// ---------------------------------------------------------------------------
__global__ __launch_bounds__(256) void k_mlp(
    const float* __restrict__ agg,     // h = x + sum(neighbors)
    const float* __restrict__ W1, const float* __restrict__ b1,
    const float* __restrict__ W2, const float* __restrict__ b2,
    float* __restrict__ out,           // pre-BN h2 written into d_out
    float* __restrict__ gstats)        // [sum:128][sumsq:128]
{
    extern __shared__ float smem[];
    float2* Wp    = (float2*)smem;          // 8192 float2
    float*  h1all = smem + 16384;
    float*  lstat = smem + 32768;

    const int tid  = threadIdx.x;
    const int wid  = tid >> 5;
    const int lane = tid & 31;
    const int half = lane >> 4;   // 0: lanes 0-15, 1: lanes 16-31
    const int l16  = lane & 15;
    const int rowBase = blockIdx.x * 128 + wid * 16;

    lstat[tid < 256 ? tid : 0] = 0.0f;

    // Stage W1 packed: Wp[kp*128+n] = { W1[2kp][n], W1[2kp+1][n] }
    for (int idx = tid; idx < 64 * 128; idx += 256) {
        int kp = idx >> 7, n = idx & 127;
        Wp[idx] = make_float2(W1[(2 * kp) * DCH + n], W1[(2 * kp + 1) * DCH + n]);
    }
    __syncthreads();

    // ---------------- GEMM1: 16x128 row stripe  x  W1 ----------------
    // A-frag layout (16x4 f32): lanes 0-15 -> K = k0,k0+1 ; lanes 16-31 -> K = k0+2,k0+3
    v8f acc[8];
#pragma unroll
    for (int n = 0; n < 8; ++n)
        acc[n] = (v8f){0.f, 0.f, 0.f, 0.f, 0.f, 0.f, 0.f, 0.f};

    {
        int arowIdx = rowBase + l16;
        if (arowIdx > N_NODES - 1) arowIdx = N_NODES - 1;   // clamp (results masked later)
        const float* arow = agg + (size_t)arowIdx * DCH;
#pragma unroll 4
        for (int k0 = 0; k0 < DCH; k0 += 4) {
            float2 af = *(const float2*)(arow + k0 + 2 * half);
            v2f a; a.x = af.x; a.y = af.y;
            const int kp = (k0 >> 1) + half;
#pragma unroll
            for (int n = 0; n < 8; ++n) {
                float2 bf = Wp[kp * DCH + n * 16 + l16];
                v2f b; b.x = bf.x; b.y = bf.y;
                acc[n] = __builtin_amdgcn_wmma_f32_16x16x4_f32(
                    false, a, false, b, (short)0, acc[n], false, false);
            }
        }
    }

    // bias1 + ReLU; flip D-layout (lane=col) -> row-major staging in LDS
    float* h1 = h1all + wid * (16 * DCH);   // wave-private 16x128 tile
#pragma unroll
    for (int n = 0; n < 8; ++n) {
        const int col = n * 16 + l16;
        const float bias = b1[col];
#pragma unroll
        for (int r = 0; r < 8; ++r) {
            float v = acc[n][r] + bias;
            v = v > 0.f ? v : 0.f;
            h1[(r + 8 * half) * DCH + col] = v;
        }
    }
    __syncthreads();                        // all waves done with W1 region

    // Stage W2 packed into the same LDS region
    for (int idx = tid; idx < 64 * 128; idx += 256) {
        int kp = idx >> 7, n = idx & 127;
        Wp[idx] = make_float2(W2[(2 * kp) * DCH + n], W2[(2 * kp + 1) * DCH + n]);
    }
    __syncthreads();

    // ---------------- GEMM2: relu(h1) x W2 ----------------
    v8f acc2[8];
#pragma unroll
    for (int n = 0; n < 8; ++n)
        acc2[n] = (v8f){0.f, 0.f, 0.f, 0.f, 0.f, 0.f, 0.f, 0.f};

#pragma unroll 4
    for (int k0 = 0; k0 < DCH; k0 += 4) {
        float2 af = *(const float2*)(h1 + l16 * DCH + k0 + 2 * half);  // ds_load_b64
        v2f a; a.x = af.x; a.y = af.y;
        const int kp = (k0 >> 1) + half;
#pragma unroll
        for (int n = 0; n < 8; ++n) {
            float2 bf = Wp[kp * DCH + n * 16 + l16];
            v2f b; b.x = bf.x; b.y = bf.y;
            acc2[n] = __builtin_amdgcn_wmma_f32_16x16x4_f32(
                false, a, false, b, (short)0, acc2[n], false, false);
        }
    }

    // bias2 + outer ReLU, store h2, accumulate BN partials
#pragma unroll
    for (int n = 0; n < 8; ++n) {
        const int col = n * 16 + l16;
        const float bias = b2[col];
        float s = 0.f, s2 = 0.f;
#pragma unroll
        for (int r = 0; r < 8; ++r) {
            const int row = rowBase + r + 8 * half;
            float v = acc2[n][r] + bias;
            v = v > 0.f ? v : 0.f;
            if (row < N_NODES) {
                out[(size_t)row * DCH + col] = v;
                s  += v;
                s2 += v * v;
            }
        }
        atomicAdd(&lstat[col], s);          // ds_add_f32
        atomicAdd(&lstat[DCH + col], s2);
    }
    __syncthreads();
    atomicAdd(&gstats[tid], lstat[tid]);    // 256 floats -> one flush per WG
}

// ---------------------------------------------------------------------------
// Kernel D: fold BN stats into per-channel scale/shift.
// ---------------------------------------------------------------------------
__global__ void k_finalize(const float* __restrict__ gstats,
                           const float* __restrict__ gamma,
                           const float* __restrict__ beta,
                           float* __restrict__ ss) {
    int c = threadIdx.x;
    if (c < DCH) {
        const float invN = 1.0f / (float)N_NODES;
        float mean = gstats[c] * invN;
        float var  = gstats[DCH + c] * invN - mean * mean;
        float inv  = rsqrtf(var + BN_EPS);
        float sc   = gamma[c] * inv;
        ss[c]       = sc;
        ss[DCH + c] = beta[c] - mean * sc;
    }
}

// ---------------------------------------------------------------------------
// Kernel E: in-place normalize d_out:  v = v*scale[c] + shift[c]  (float4).
// ---------------------------------------------------------------------------
__global__ void k_bn(float* __restrict__ outp, const float* __restrict__ ss, int n4) {
    int i = blockIdx.x * blockDim.x + threadIdx.x;
    if (i >= n4) return;
    float4* out4 = (float4*)outp;
    const int c0 = (i * 4) & (DCH - 1);
    float4 v  = out4[i];
    float4 sc = *(const float4*)(ss + c0);
    float4 sh = *(const float4*)(ss + DCH + c0);
    v.x = v.x * sc.x + sh.x;
    v.y = v.y * sc.y + sh.y;
    v.z = v.z * sc.z + sh.z;
    v.w = v.w * sc.w + sh.w;
    out4[i] = v;
}

// ---------------------------------------------------------------------------
extern "C" void kernel_launch(void* const* d_in, const int* in_sizes, int n_in,
                              void* d_out, int out_size, void* d_ws, size_t ws_size,
                              hipStream_t stream) {
    const float* x     = (const float*)d_in[0];
    const int*   ei    = (const int*)  d_in[1];
    const float* W1    = (const float*)d_in[2];
    const float* b1    = (const float*)d_in[3];
    const float* W2    = (const float*)d_in[4];
    const float* b2    = (const float*)d_in[5];
    const float* gamma = (const float*)d_in[6];
    const float* beta  = (const float*)d_in[7];
    float*       out   = (float*)d_out;

    // workspace: agg [N*D] | gstats [256] | scale/shift [256]
    float* ws     = (float*)d_ws;
    float* agg    = ws;
    float* gstats = ws + (size_t)N_NODES * DCH;
    float* ss     = gstats + 2 * DCH;

    const int n4 = (N_NODES * DCH) / 4;   // 3,200,000 float4s

    // A: agg = x, zero stats
    k_init<<<(n4 + 255) / 256, 256, 0, stream>>>(
        (const float4*)x, (float4*)agg, gstats, n4);

    // B: scatter-add edges (one wave per edge)
    {
        const int wavesPerBlock = 256 / 32;
        const int blocks = (N_EDGES + wavesPerBlock - 1) / wavesPerBlock;
        k_scatter<<<blocks, 256, 0, stream>>>(x, ei, agg, N_EDGES);
    }

    // C: fused MLP (2x WMMA f32 GEMM) + BN partial stats
    {
        const int blocks = (N_NODES + 127) / 128;   // 782
        const size_t lds = (size_t)(16384 + 16384 + 256) * sizeof(float); // 132096 B
        k_mlp<<<blocks, 256, lds, stream>>>(agg, W1, b1, W2, b2, out, gstats);
    }

    // D: stats -> scale/shift
    k_finalize<<<1, 128, 0, stream>>>(gstats, gamma, beta, ss);

    // E: in-place normalize
    k_bn<<<(n4 + 255) / 256, 256, 0, stream>>>(out, ss, n4);
}